// LightGCN_46480136077527
// MI455X (gfx1250) — compile-verified
//
#include <hip/hip_runtime.h>

// LightGCN on MI455X (gfx1250, wave32).
//
// Memory-bound pipeline (~300 MB HBM traffic -> ~12us at 23.3 TB/s). The
// 38.4MB embedding table is L2-resident (192MB L2), so x[col] gathers hit L2.
// SpMM uses V_WMMA_F32_16X16X4_F32 as a scatter-accumulate engine over CSR
// row-tiles of 16 rows per wave: no atomics, no zero-init of outputs, and the
// "acc += layer" update is fused into the C-tile flush.
//
// One wave per workgroup in the SpMM kernel so that r0/e0/e1/loop counter are
// compiler-provably uniform: edge fetches lower to s_load through the scalar
// cache, EXEC stays all-ones for WMMA, and the steady-state loop is split from
// a masked epilogue so the hot path has zero clamping/branching per group.

typedef __attribute__((ext_vector_type(2))) float v2f;
typedef __attribute__((ext_vector_type(8))) float v8f;

#define LATENT 64

// ---------------------------------------------------------------- row_ptr ---
__global__ __launch_bounds__(256)
void lgcn_rowptr_kernel(const int* __restrict__ rows, int nnz, int nnodes,
                        int* __restrict__ row_ptr) {
  int r = blockIdx.x * blockDim.x + threadIdx.x;
  if (r > nnodes) return;
  int lo = 0, hi = nnz;                       // lower_bound(rows, r)
  while (lo < hi) {
    int mid = (lo + hi) >> 1;
    if (rows[mid] < r) lo = mid + 1; else hi = mid;
  }
  row_ptr[r] = lo;
}

// --------------------------------------------------- concat + seed acc ------
__global__ __launch_bounds__(256)
void lgcn_init_kernel(const float4* __restrict__ u4, const float4* __restrict__ i4,
                      float4* __restrict__ cur, float4* __restrict__ acc,
                      int user4, int total4) {
  int stride = gridDim.x * blockDim.x;
  for (int idx = blockIdx.x * blockDim.x + threadIdx.x; idx < total4; idx += stride) {
    float4 v = (idx < user4) ? u4[idx] : i4[idx - user4];
    cur[idx] = v;
    acc[idx] = v;
  }
}

// -------------------------------------- 4-edge group -> 4 WMMAs (16x16x4) ---
__device__ __forceinline__
void lgcn_group4(int r0, int m, int half,
                 int ra0, int ra1, int ra2, int ra3,
                 int ca0, int ca1, int ca2, int ca3,
                 float va0, float va1, float va2, float va3,
                 const float* __restrict__ x,
                 v8f& c0, v8f& c1, v8f& c2, v8f& c3) {
  // This lane's two K slots: k0 = 2*half, k1 = 2*half+1 (ISA A/B layouts).
  int   rk0 = half ? ra2 : ra0,  rk1 = half ? ra3 : ra1;
  int   ck0 = half ? ca2 : ca0,  ck1 = half ? ca3 : ca1;
  float vk0 = half ? va2 : va0,  vk1 = half ? va3 : va1;

  // A[m][k] = val_k iff edge k scatters to local row m (else 0).
  v2f a;
  a.x = (rk0 - r0 == m) ? vk0 : 0.f;
  a.y = (rk1 - r0 == m) ? vk1 : 0.f;

  // B[k][n] gather (L2 hits). 32-bit offsets -> saddr + voffset loads.
  unsigned off0 = ((unsigned)ck0 << 6) + (unsigned)m;
  unsigned off1 = ((unsigned)ck1 << 6) + (unsigned)m;
  v2f b0, b1, b2, b3;
  b0.x = x[off0];        b0.y = x[off1];
  b1.x = x[off0 + 16];   b1.y = x[off1 + 16];
  b2.x = x[off0 + 32];   b2.y = x[off1 + 32];
  b3.x = x[off0 + 48];   b3.y = x[off1 + 48];

  c0 = __builtin_amdgcn_wmma_f32_16x16x4_f32(false, a, false, b0, (short)0, c0, false, false);
  c1 = __builtin_amdgcn_wmma_f32_16x16x4_f32(false, a, false, b1, (short)0, c1, false, false);
  c2 = __builtin_amdgcn_wmma_f32_16x16x4_f32(false, a, false, b2, (short)0, c2, false, false);
  c3 = __builtin_amdgcn_wmma_f32_16x16x4_f32(false, a, false, b3, (short)0, c3, false, false);
}

// ------------------------------------------- SpMM via WMMA f32 16x16x4 ------
// One wave (= one workgroup) per 16-row tile.
// mode bit0: write `out`; bit1: accumulate `acc`.
__global__ __launch_bounds__(32)
void lgcn_spmm_wmma_kernel(const int* __restrict__ row_ptr,
                           const int* __restrict__ erows,
                           const int* __restrict__ ecols,
                           const float* __restrict__ evals,
                           const float* __restrict__ x,
                           float* __restrict__ out,
                           float* __restrict__ acc,
                           int nnodes, int mode) {
  const int lane = threadIdx.x;           // 0..31
  const int r0   = blockIdx.x << 4;       // uniform: 16-row tile base
  if (r0 >= nnodes) return;

  const int m    = lane & 15;   // A: row / B,C: column within tile
  const int half = lane >> 4;   // K half: lanes 0-15 -> K=0,1; 16-31 -> K=2,3

  int rend = r0 + 16; if (rend > nnodes) rend = nnodes;
  const int e0 = row_ptr[r0];             // uniform -> s_load
  const int e1 = row_ptr[rend];

  v8f c0 = {}, c1 = {}, c2 = {}, c3 = {};   // 16x16 C tiles, features 0..63

  // ---- steady state: 8 edges (2 groups, 8 WMMAs) per iteration, no masks ---
  const int eFull = e0 + ((e1 - e0) & ~7);
  int e = e0;
  for (; e < eFull; e += 8) {
    if (((e - e0) & 24) == 0 && e + 256 < e1) {   // uniform, every 4th iter
      __builtin_prefetch(erows + e + 256, 0, 1);
      __builtin_prefetch(ecols + e + 256, 0, 1);
      __builtin_prefetch(evals + e + 256, 0, 1);
    }
    lgcn_group4(r0, m, half,
                erows[e],   erows[e+1], erows[e+2], erows[e+3],
                ecols[e],   ecols[e+1], ecols[e+2], ecols[e+3],
                evals[e],   evals[e+1], evals[e+2], evals[e+3],
                x, c0, c1, c2, c3);
    lgcn_group4(r0, m, half,
                erows[e+4], erows[e+5], erows[e+6], erows[e+7],
                ecols[e+4], ecols[e+5], ecols[e+6], ecols[e+7],
                evals[e+4], evals[e+5], evals[e+6], evals[e+7],
                x, c0, c1, c2, c3);
  }

  // ---- epilogue: up to 7 edges, clamped indices + value masking -----------
  const int eLast = e1 - 1;
  for (; e < e1; e += 4) {
    int i1 = e + 1 > eLast ? eLast : e + 1;
    int i2 = e + 2 > eLast ? eLast : e + 2;
    int i3 = e + 3 > eLast ? eLast : e + 3;
    float va1 = (e + 1 <= eLast) ? evals[i1] : 0.f;
    float va2 = (e + 2 <= eLast) ? evals[i2] : 0.f;
    float va3 = (e + 3 <= eLast) ? evals[i3] : 0.f;
    lgcn_group4(r0, m, half,
                erows[e], erows[i1], erows[i2], erows[i3],
                ecols[e], ecols[i1], ecols[i2], ecols[i3],
                evals[e], va1, va2, va3,
                x, c0, c1, c2, c3);
  }

  // Flush: VGPR j of C holds M=j (lanes 0-15) / M=j+8 (lanes 16-31), N=m.
  // Rows are exclusively owned by this wave -> plain stores / RMW, no atomics.
#pragma unroll
  for (int j = 0; j < 8; ++j) {
    int row = r0 + j + 8 * half;
    if (row < nnodes) {
      size_t base = (size_t)row * LATENT + m;
      if (mode & 1) {
        out[base]      = c0[j];
        out[base + 16] = c1[j];
        out[base + 32] = c2[j];
        out[base + 48] = c3[j];
      }
      if (mode & 2) {
        acc[base]      += c0[j];
        acc[base + 16] += c1[j];
        acc[base + 32] += c2[j];
        acc[base + 48] += c3[j];
      }
    }
  }
}

// ----------------------------------------------- batched dot (gamma) --------
__global__ __launch_bounds__(256)
void lgcn_gamma_kernel(const float* __restrict__ acc,
                       const int* __restrict__ users, const int* __restrict__ items,
                       float* __restrict__ out, int numUsers, int batch) {
  int wave = (blockIdx.x * blockDim.x + threadIdx.x) >> 5;
  int lane = threadIdx.x & 31;
  if (wave >= batch) return;
  int u  = users[wave];
  int it = items[wave];
  const float2* up = (const float2*)(acc + (size_t)u * LATENT);
  const float2* ip = (const float2*)(acc + (size_t)(numUsers + it) * LATENT);
  float2 a = up[lane], b = ip[lane];
  float p = a.x * b.x + a.y * b.y;
#pragma unroll
  for (int off = 16; off; off >>= 1) p += __shfl_xor(p, off, 32);
  // light_out = acc/4 -> gamma = dot(acc_u, acc_i) / 16
  if (lane == 0) out[wave] = p * 0.0625f;
}

// ---------------------------------------------------------------- launch ----
extern "C" void kernel_launch(void* const* d_in, const int* in_sizes, int n_in,
                              void* d_out, int out_size, void* d_ws, size_t ws_size,
                              hipStream_t stream) {
  const float* user_emb = (const float*)d_in[0];
  const float* item_emb = (const float*)d_in[1];
  const int*   erows    = (const int*)d_in[2];
  const int*   ecols    = (const int*)d_in[3];
  const float* evals    = (const float*)d_in[4];
  const int*   users    = (const int*)d_in[5];
  const int*   items    = (const int*)d_in[6];

  const int numUsers = in_sizes[0] / LATENT;
  const int numItems = in_sizes[1] / LATENT;
  const int nnodes   = numUsers + numItems;
  const int nnz      = in_sizes[2];
  const int batch    = in_sizes[5];

  // Workspace layout: row_ptr | buf0 | buf1 | buf2 | acc  (~155 MB total)
  char*  ws       = (char*)d_ws;
  size_t embBytes = (size_t)nnodes * LATENT * sizeof(float);
  int*   row_ptr  = (int*)ws;  ws += (((size_t)(nnodes + 1) * 4) + 255) & ~(size_t)255;
  float* buf0     = (float*)ws; ws += embBytes;
  float* buf1     = (float*)ws; ws += embBytes;
  float* buf2     = (float*)ws; ws += embBytes;
  float* acc      = (float*)ws;

  // 1) CSR row pointers from sorted edge_rows.
  lgcn_rowptr_kernel<<<(nnodes + 1 + 255) / 256, 256, 0, stream>>>(
      erows, nnz, nnodes, row_ptr);

  // 2) cur = concat(user_emb, item_emb); acc = cur.
  int total4 = nnodes * (LATENT / 4);
  int user4  = numUsers * (LATENT / 4);
  int initBlocks = (total4 + 255) / 256; if (initBlocks > 4096) initBlocks = 4096;
  lgcn_init_kernel<<<initBlocks, 256, 0, stream>>>(
      (const float4*)user_emb, (const float4*)item_emb,
      (float4*)buf0, (float4*)acc, user4, total4);

  // 3) Three propagation layers; each fuses acc += A@cur into the flush.
  int tiles = (nnodes + 15) / 16;       // one single-wave workgroup per tile
  lgcn_spmm_wmma_kernel<<<tiles, 32, 0, stream>>>(
      row_ptr, erows, ecols, evals, buf0, buf1, acc, nnodes, /*out+acc*/3);
  lgcn_spmm_wmma_kernel<<<tiles, 32, 0, stream>>>(
      row_ptr, erows, ecols, evals, buf1, buf2, acc, nnodes, /*out+acc*/3);
  lgcn_spmm_wmma_kernel<<<tiles, 32, 0, stream>>>(
      row_ptr, erows, ecols, evals, buf2, buf1, acc, nnodes, /*acc only*/2);

  // 4) gamma[b] = dot(light_out_u, light_out_i).
  lgcn_gamma_kernel<<<(batch + 7) / 8, 256, 0, stream>>>(
      acc, users, items, (float*)d_out, numUsers, batch);
}